// HamiltonianPotential_35759897706556
// MI455X (gfx1250) — compile-verified
//
#include <hip/hip_runtime.h>
#include <math.h>

#define BB 2
#define NN 128
#define KK 32
#define VOCAB 32000
#define EPSF 1e-8f

typedef float v2f __attribute__((ext_vector_type(2)));
typedef float v8f __attribute__((ext_vector_type(8)));

// ---------------------------------------------------------------------------
// Wave-cooperative 32x32x32 fp32 matmul on LDS: C = op(A) * B.
// Uses V_WMMA_F32_16X16X4_F32: 4 waves (128 threads), one 16x16 tile each,
// 8 k-steps of K=4. Operand lane layout per CDNA5 ISA:
//   A (16x4):  lane m (0-15) half h: a[v] = A[m, k0 + 2h + v]
//   B (4x16):  lane n (0-15) half h: b[v] = B[k0 + 2h + v, n]
//   C (16x16): lanes 0-15 N=lane M=r; lanes 16-31 N=lane-16 M=8+r
// C must not alias A or B.
// ---------------------------------------------------------------------------
__device__ inline void wave_mm(float* C, const float* A, const float* B, bool transA) {
  __syncthreads();                       // operands ready
  const int tid  = threadIdx.x;
  const int wave = tid >> 5, lane = tid & 31;
  const int half = lane >> 4, m16 = lane & 15;
  const int tm = (wave >> 1) << 4, tn = (wave & 1) << 4;
  v8f acc = {0.f, 0.f, 0.f, 0.f, 0.f, 0.f, 0.f, 0.f};
#pragma unroll
  for (int k0 = 0; k0 < 32; k0 += 4) {
    const int ka = k0 + 2 * half;
    v2f av, bv;
    if (!transA) {
      av.x = A[(tm + m16) * 32 + ka];
      av.y = A[(tm + m16) * 32 + ka + 1];
    } else {
      av.x = A[ka * 32 + (tm + m16)];
      av.y = A[(ka + 1) * 32 + (tm + m16)];
    }
    bv.x = B[ka * 32 + tn + m16];
    bv.y = B[(ka + 1) * 32 + tn + m16];
    acc = __builtin_amdgcn_wmma_f32_16x16x4_f32(false, av, false, bv,
                                                (short)0, acc, false, false);
  }
#pragma unroll
  for (int r = 0; r < 8; ++r)
    C[(tm + 8 * half + r) * 32 + tn + m16] = acc[r];
  __syncthreads();                       // result visible
}

// In-place Cholesky of SPD 32x32 in LDS (lower + diag become L).
// Returns logdet = 2*sum(log diag(L)). Needs red[33] scratch. 128 threads.
__device__ inline float chol_logdet(float* M, float* red) {
  const int tid = threadIdx.x;
  for (int k = 0; k < 32; ++k) {
    if (tid == 0) M[k * 33] = sqrtf(M[k * 33]);
    __syncthreads();
    const float dk = M[k * 33];
    if (tid > k && tid < 32) M[tid * 32 + k] /= dk;
    __syncthreads();
    for (int e = tid; e < 1024; e += 128) {
      const int i = e >> 5, j = e & 31;
      if (i > k && j > k && j <= i) M[e] -= M[i * 32 + k] * M[j * 32 + k];
    }
    __syncthreads();
  }
  if (tid < 32) red[tid] = logf(M[tid * 33]);
  __syncthreads();
  if (tid == 0) {
    float s = 0.f;
    for (int i = 0; i < 32; ++i) s += red[i];
    red[32] = 2.f * s;
  }
  __syncthreads();
  return red[32];
}

// Y = L^{-1} (forward substitution, one column per thread). Y in LDS.
__device__ inline void trisolve_inv(const float* L, float* Y) {
  const int tid = threadIdx.x;
  if (tid < 32) {
    const int c = tid;
    for (int i = 0; i < 32; ++i) {
      if (i < c) { Y[i * 32 + c] = 0.f; continue; }
      float s = (i == c) ? 1.f : 0.f;
      for (int j = c; j < i; ++j) s -= L[i * 32 + j] * Y[j * 32 + c];
      Y[i * 32 + c] = s / L[i * 33];
    }
  }
  __syncthreads();
}

// ---------------------------------------------------------------------------
// Kernel 1: per-node features + self-KL. One block (4 waves) per (b,n).
// ---------------------------------------------------------------------------
__global__ __launch_bounds__(128)
void node_kernel(const float* __restrict__ mu, const float* __restrict__ Sigma,
                 const float* __restrict__ phi, const float* __restrict__ mu_prior,
                 const float* __restrict__ Sigma_prior, const float* __restrict__ gen,
                 float* __restrict__ Qt, float* __restrict__ Xf,
                 float* __restrict__ vbuf, float* __restrict__ wt,
                 float* __restrict__ cval, float* __restrict__ ldb,
                 float* __restrict__ acc) {
  __shared__ float sG[3 * 1024];
  __shared__ float sM0[1024], sM1[1024], sM2[1024], sM3[1024], sM4[1024];
  __shared__ float s_mu[32], s_mup[32], s_v[32], s_pmu[32], s_w[32], s_d[32], s_t[32];
  __shared__ float s_red[136];
  __shared__ float s_sc[2];

  const int tid  = threadIdx.x;
  const int node = blockIdx.x;              // 0..B*N-1
  const int b = node >> 7, n = node & 127;
  const size_t nb = (size_t)node;
  const size_t mb = nb * 1024;

  for (int e = tid; e < 3072; e += 128) sG[e] = gen[e];
  if (tid < 32) {
    s_mu[tid]  = mu[nb * 32 + tid];
    s_mup[tid] = mu_prior[nb * 32 + tid];
  }
  const float p0 = phi[nb * 3 + 0], p1 = phi[nb * 3 + 1], p2 = phi[nb * 3 + 2];
  __syncthreads();

  // A_s = (sum_a phi_a G_a) / 32  (scaling for squaring, s=5)
  for (int e = tid; e < 1024; e += 128)
    sM0[e] = (p0 * sG[e] + p1 * sG[1024 + e] + p2 * sG[2048 + e]) * (1.f / 32.f);
  // T = I
  for (int e = tid; e < 1024; e += 128) sM1[e] = ((e >> 5) == (e & 31)) ? 1.f : 0.f;
  __syncthreads();

  // expm: Horner Taylor-10 then 5 squarings (all WMMA matmuls)
  for (int k = 10; k >= 1; --k) {
    wave_mm(sM2, sM0, sM1, false);          // TMP = A_s * T
    const float rk = 1.f / (float)k;
    for (int e = tid; e < 1024; e += 128)
      sM1[e] = sM2[e] * rk + (((e >> 5) == (e & 31)) ? 1.f : 0.f);
    __syncthreads();
  }
  float* cur = sM1;
  float* oth = sM2;
  for (int t = 0; t < 5; ++t) {
    wave_mm(oth, cur, cur, false);
    float* tp = cur; cur = oth; oth = tp;
  }
  float* E  = cur;   // exp(phiM)  (== sM2)
  float* F1 = oth;   // free scratch (== sM1)

  // SQ = Sigma + eps*I
  for (int e = tid; e < 1024; e += 128)
    sM0[e] = Sigma[mb + e] + (((e >> 5) == (e & 31)) ? EPSF : 0.f);
  __syncthreads();
  for (int e = tid; e < 1024; e += 128) F1[e] = sM0[e];
  __syncthreads();
  const float ld_i = chol_logdet(F1, s_red);
  trisolve_inv(F1, sM3);                    // Y = L^{-1}
  wave_mm(sM4, sM3, sM3, true);             // P = Y^T Y = SQ^{-1}

  // vectors: v = E^T mu, Pmu = P mu, w = E^T Pmu, c = mu^T Pmu
  if (tid < 32) {
    float sv = 0.f, sp = 0.f;
    for (int m = 0; m < 32; ++m) sv += E[m * 32 + tid] * s_mu[m];
    for (int j = 0; j < 32; ++j) sp += sM4[tid * 32 + j] * s_mu[j];
    s_v[tid] = sv;
    s_pmu[tid] = sp;
  }
  __syncthreads();
  if (tid < 32) {
    float sw = 0.f;
    for (int m = 0; m < 32; ++m) sw += E[m * 32 + tid] * s_pmu[m];
    s_w[tid] = sw;
  }
  if (tid == 0) {
    float cc = 0.f;
    for (int m = 0; m < 32; ++m) cc += s_mu[m] * s_pmu[m];
    s_sc[0] = cc;
  }
  __syncthreads();

  // Q = E^T (P E)  -> write f-major (transposed across nodes) for Gram GEMM
  wave_mm(F1, sM4, E, false);
  wave_mm(sM3, E, F1, true);
  for (int e = tid; e < 1024; e += 128)
    Qt[(size_t)b * 131072 + (size_t)e * 128 + n] = sM3[e];

  // X = E^T (SQ E) + v v^T  -> row-major per node
  wave_mm(F1, sM0, E, false);
  wave_mm(sM3, E, F1, true);
  for (int e = tid; e < 1024; e += 128)
    Xf[mb + e] = sM3[e] + s_v[e >> 5] * s_v[e & 31];

  if (tid < 32) {
    vbuf[nb * 32 + tid] = s_v[tid];
    wt[(size_t)b * 4096 + (size_t)tid * 128 + n] = s_w[tid];
  }
  if (tid == 0) { cval[node] = s_sc[0]; ldb[node] = ld_i; }
  __syncthreads();

  // ---- self KL: KL(N(mu,SQ) || N(mu_p, Sp)) ----
  for (int e = tid; e < 1024; e += 128)
    F1[e] = Sigma_prior[mb + e] + (((e >> 5) == (e & 31)) ? EPSF : 0.f);
  __syncthreads();
  const float ld_p = chol_logdet(F1, s_red);
  trisolve_inv(F1, sM3);
  wave_mm(sM4, sM3, sM3, true);             // Pp = Sp^{-1}
  float part = 0.f;
  for (int e = tid; e < 1024; e += 128) part += sM4[e] * sM0[e];  // tr(Pp SQ)
  s_red[tid] = part;
  __syncthreads();
  for (int st = 64; st > 0; st >>= 1) {
    if (tid < st) s_red[tid] += s_red[tid + st];
    __syncthreads();
  }
  if (tid < 32) s_d[tid] = s_mup[tid] - s_mu[tid];
  __syncthreads();
  if (tid < 32) {
    float tt = 0.f;
    for (int j = 0; j < 32; ++j) tt += sM4[tid * 32 + j] * s_d[j];
    s_t[tid] = tt;
  }
  __syncthreads();
  if (tid == 0) {
    float mah = 0.f;
    for (int m = 0; m < 32; ++m) mah += s_d[m] * s_t[m];
    const float kl = 0.5f * (s_red[0] + mah - 32.f + ld_p - ld_i);
    atomicAdd(&acc[b], /*ALPHA*/ 1.0f * kl);   // self accumulator
  }
}

// ---------------------------------------------------------------------------
// Kernel 2: pairwise alignment term via two Gram GEMMs.
// 8 waves/block; each wave owns one 16x16 (i,j) tile of the 128x128 matrix.
// kl_ij = 0.5*(<Q_j,X_i> + c_j - 2<w_j,v_i> - K + ld_j - ld_i)
// ---------------------------------------------------------------------------
__global__ __launch_bounds__(256)
void align_kernel(const float* __restrict__ beta, const float* __restrict__ Qt,
                  const float* __restrict__ Xf, const float* __restrict__ vbuf,
                  const float* __restrict__ wt, const float* __restrict__ cval,
                  const float* __restrict__ ldb, float* __restrict__ acc) {
  const int tid  = threadIdx.x;
  const int wave = tid >> 5, lane = tid & 31;
  const int half = lane >> 4, m16 = lane & 15;
  const int b    = blockIdx.x >> 3;
  const int tile = ((blockIdx.x & 7) << 3) + wave;    // 0..63
  const int ti = tile >> 3, tj = tile & 7;
  const int i_base = ti * 16, j_base = tj * 16;

  const float* Xr  = Xf + (size_t)b * 131072 + (size_t)(i_base + m16) * 1024;
  const float* Qtb = Qt + (size_t)b * 131072;
  const float* vr  = vbuf + (size_t)b * 4096 + (size_t)(i_base + m16) * 32;
  const float* wtb = wt + (size_t)b * 4096;

  v8f acc1 = {0.f, 0.f, 0.f, 0.f, 0.f, 0.f, 0.f, 0.f};
  for (int f0 = 0; f0 < 1024; f0 += 4) {              // <Q_j, X_i> over F=1024
    const int ka = f0 + 2 * half;
    v2f av, bv;
    av.x = Xr[ka];
    av.y = Xr[ka + 1];
    bv.x = Qtb[(size_t)ka * 128 + j_base + m16];
    bv.y = Qtb[(size_t)(ka + 1) * 128 + j_base + m16];
    acc1 = __builtin_amdgcn_wmma_f32_16x16x4_f32(false, av, false, bv,
                                                 (short)0, acc1, false, false);
  }
  v8f acc2 = {0.f, 0.f, 0.f, 0.f, 0.f, 0.f, 0.f, 0.f};
  for (int k0 = 0; k0 < 32; k0 += 4) {                // <w_j, v_i> over K=32
    const int ka = k0 + 2 * half;
    v2f av, bv;
    av.x = vr[ka];
    av.y = vr[ka + 1];
    bv.x = wtb[ka * 128 + j_base + m16];
    bv.y = wtb[(ka + 1) * 128 + j_base + m16];
    acc2 = __builtin_amdgcn_wmma_f32_16x16x4_f32(false, av, false, bv,
                                                 (short)0, acc2, false, false);
  }

  const int j = j_base + m16;
  const float cj  = cval[b * 128 + j];
  const float ldj = ldb[b * 128 + j];
  float sum = 0.f;
#pragma unroll
  for (int r = 0; r < 8; ++r) {
    const int i = i_base + 8 * half + r;
    const float ldi = ldb[b * 128 + i];
    const float kl = 0.5f * (acc1[r] + cj - 2.f * acc2[r] - 32.f + ldj - ldi);
    if (i != j) sum += beta[((size_t)b * 128 + i) * 128 + j] * kl;
  }
  atomicAdd(&acc[2 + b], /*LAMBDA*/ 1.0f * sum);       // align accumulator
}

// ---------------------------------------------------------------------------
// Kernel 3: cross-entropy, one block per (b,n), online softmax over VOCAB.
// ---------------------------------------------------------------------------
__global__ __launch_bounds__(256)
void ce_kernel(const float* __restrict__ mu, const int* __restrict__ targets,
               const float* __restrict__ W_out, float* __restrict__ acc) {
  __shared__ __align__(16) float s_mu[32];
  __shared__ float s_m[256], s_s[256];
  __shared__ float s_tl;

  const int tid = threadIdx.x;
  const int b = blockIdx.x / NN, n = blockIdx.x % NN;
  if (tid < 32) s_mu[tid] = mu[((size_t)b * NN + n) * 32 + tid];
  const int target = targets[b * NN + n];
  __syncthreads();

  float mloc = -3.4e38f, sloc = 0.f;
  for (int v = tid; v < VOCAB; v += 256) {
    const float4* wr = (const float4*)(W_out + (size_t)v * 32);
    float dot = 0.f;
#pragma unroll
    for (int q = 0; q < 8; ++q) {
      const float4 wv4 = wr[q];
      const float4 mv  = ((const float4*)s_mu)[q];
      dot += wv4.x * mv.x + wv4.y * mv.y + wv4.z * mv.z + wv4.w * mv.w;
    }
    if (v == target) s_tl = dot;
    if (dot > mloc) { sloc = sloc * expf(mloc - dot) + 1.f; mloc = dot; }
    else            { sloc += expf(dot - mloc); }
  }
  s_m[tid] = mloc;
  s_s[tid] = sloc;
  __syncthreads();
  for (int st = 128; st > 0; st >>= 1) {
    if (tid < st) {
      const float m1 = s_m[tid], s1 = s_s[tid];
      const float m2 = s_m[tid + st], s2 = s_s[tid + st];
      const float M = fmaxf(m1, m2);
      s_s[tid] = s1 * expf(m1 - M) + s2 * expf(m2 - M);
      s_m[tid] = M;
    }
    __syncthreads();
  }
  if (tid == 0) {
    const float ce = (s_m[0] + logf(s_s[0])) - s_tl;   // logsumexp - logit_target
    atomicAdd(&acc[4 + b], ce);
  }
}

__global__ void zero_kernel(float* acc) {
  if (threadIdx.x < 8) acc[threadIdx.x] = 0.f;
}

__global__ void final_kernel(const float* __restrict__ acc, float* __restrict__ out) {
  const int b = threadIdx.x;
  if (b < BB) out[b] = acc[b] + acc[2 + b] + acc[4 + b];
}

// ---------------------------------------------------------------------------
extern "C" void kernel_launch(void* const* d_in, const int* in_sizes, int n_in,
                              void* d_out, int out_size, void* d_ws, size_t ws_size,
                              hipStream_t stream) {
  (void)in_sizes; (void)n_in; (void)out_size; (void)ws_size;
  const float* mu          = (const float*)d_in[0];
  const float* Sigma       = (const float*)d_in[1];
  const float* phi         = (const float*)d_in[2];
  const float* mu_prior    = (const float*)d_in[3];
  const float* Sigma_prior = (const float*)d_in[4];
  const float* beta        = (const float*)d_in[5];
  const int*   targets     = (const int*)d_in[6];
  const float* W_out       = (const float*)d_in[7];
  const float* generators  = (const float*)d_in[8];
  float* out = (float*)d_out;

  // workspace layout (floats): ~2.1 MB total
  float* ws   = (float*)d_ws;
  float* Qt   = ws;                 // B*1024*N          = 262144
  float* Xf   = Qt + 262144;        // B*N*1024          = 262144
  float* vbuf = Xf + 262144;        // B*N*K             = 8192
  float* wt   = vbuf + 8192;        // B*K*N             = 8192
  float* cvl  = wt + 8192;          // B*N               = 256
  float* ldb  = cvl + 256;          // B*N               = 256
  float* acc  = ldb + 256;          // [self0,self1, al0,al1, ce0,ce1, pad]

  zero_kernel<<<1, 32, 0, stream>>>(acc);
  node_kernel<<<BB * NN, 128, 0, stream>>>(mu, Sigma, phi, mu_prior, Sigma_prior,
                                           generators, Qt, Xf, vbuf, wt, cvl, ldb, acc);
  align_kernel<<<BB * 8, 256, 0, stream>>>(beta, Qt, Xf, vbuf, wt, cvl, ldb, acc);
  ce_kernel<<<BB * NN, 256, 0, stream>>>(mu, targets, W_out, acc);
  final_kernel<<<1, 32, 0, stream>>>(acc, out);
}